// SelfAttention_v1_10428180595252
// MI455X (gfx1250) — compile-verified
//
#include <hip/hip_runtime.h>

#define SEQ 4096
#define DIM 1024

typedef __attribute__((ext_vector_type(2))) float v2f;
typedef __attribute__((ext_vector_type(8))) float v8f;
typedef __attribute__((ext_vector_type(4))) unsigned u32x4;
typedef __attribute__((ext_vector_type(4))) int i32x4;
typedef __attribute__((ext_vector_type(8))) int i32x8;

#define XSTRIDE 36  // 32 data + 4 pad DWORDs (conflict-free A-frag b64 reads)
#define WSTRIDE 80  // 64 data + 16 pad DWORDs (half-waves hit disjoint bank halves)

// D(16x16,f32) = A(16x4,f32) * B(4x16,f32) + C   -> v_wmma_f32_16x16x4_f32
static __device__ __forceinline__ v8f wmma_f32_k4(v2f a, v2f b, v8f c) {
  return __builtin_amdgcn_wmma_f32_16x16x4_f32(false, a, false, b, (short)0, c,
                                               false, false);
}

#if __has_builtin(__builtin_amdgcn_tensor_load_to_lds) && \
    __has_builtin(__builtin_amdgcn_s_wait_tensorcnt)
#define USE_TDM 1
#else
#define USE_TDM 0
#endif

static __device__ __forceinline__ unsigned ldsOff(const void* p) {
  // generic (flat) LDS pointer: low 32 bits are the in-aperture LDS byte address
  return (unsigned)(unsigned long long)p;
}

#if USE_TDM
// Issue one TDM 2D tile copy global->LDS. tileW/tileH in elements (f32),
// rowStride = global row pitch in elements. LDS row pitch = tileW + pad,
// expressed via D# pad_interval/pad_amount codes:
//   pad fires when bytesStored >= (8 << padIntervalCode); adds (padAmountCode+1) DWORDs.
static __device__ __forceinline__ void tdm_load_2d(unsigned ldsByteOff,
                                                   const float* g, unsigned tileW,
                                                   unsigned tileH,
                                                   unsigned rowStride,
                                                   unsigned padIntervalCode,
                                                   unsigned padAmountCode) {
  const unsigned long long ga = (unsigned long long)(const void*)g;
  u32x4 g0;
  g0.x = 1u;                   // count=1 (valid user descriptor), no gather
  g0.y = ldsByteOff;           // lds_addr
  g0.z = (unsigned)ga;         // global_addr[31:0]
  g0.w = (unsigned)((ga >> 32) & 0x1ffffffu) | 0x80000000u;  // addr[56:32] | type=2
  i32x8 g1;
  g1[0] = (int)((2u << 16)                     // data_size = 4 bytes
                | (1u << 20)                   // pad_enable
                | (padIntervalCode << 22) | (padAmountCode << 25));
  g1[1] = (int)((tileW & 0xffffu) << 16);                                 // tensor_dim0[15:0]
  g1[2] = (int)(((tileW >> 16) & 0xffffu) | ((tileH & 0xffffu) << 16));   // dim0 hi | dim1 lo
  g1[3] = (int)(((tileH >> 16) & 0xffffu) | ((tileW & 0xffffu) << 16));   // dim1 hi | tile_dim0
  g1[4] = (int)(tileH & 0xffffu);                                         // tile_dim1, tile_dim2=0
  g1[5] = (int)rowStride;                                                 // tensor_dim0_stride lo
  g1[6] = 0;                                                              // stride hi | dim1_stride lo
  g1[7] = 0;
  const i32x4 z4 = {0, 0, 0, 0};
#if __clang_major__ >= 23
  const i32x8 z8 = {0, 0, 0, 0, 0, 0, 0, 0};
  __builtin_amdgcn_tensor_load_to_lds(g0, g1, z4, z4, z8, 0);
#else
  __builtin_amdgcn_tensor_load_to_lds(g0, g1, z4, z4, 0);
#endif
}
#endif

// ---------------------------------------------------------------------------
// Out[64x64] tile of X[SEQ x DIM] * W[DIM x DIM]; 4 waves, 32x32 per wave.
// Double-buffered LDS: next k-slab staged (TDM by wave 0, or batched b128
// register path) while all waves run f32 WMMAs on the current slab.
// X staged [m][k] stride 36; W staged row-major [k][n] stride 80.
// transposeOut=1 writes Out[n*SEQ+m] (produces V^T for the PV GEMM).
// ---------------------------------------------------------------------------
__global__ __launch_bounds__(128) void qkv_gemm(const float* __restrict__ X,
                                                const float* __restrict__ W,
                                                float* __restrict__ Out,
                                                int transposeOut) {
  __shared__ __align__(16) float sX[2][64 * XSTRIDE];
  __shared__ __align__(16) float sW[2][32 * WSTRIDE];

  const int t = threadIdx.x;
  const int lane = t & 31;
  const int wave = t >> 5;
  const int wm = (wave >> 1) << 5;  // 0 / 32
  const int wn = (wave & 1) << 5;   // 0 / 32
  const int row0 = blockIdx.y << 6;
  const int col0 = blockIdx.x << 6;
  const int lr = lane & 15;
  const int hi = lane >> 4;
  const int ksel = hi << 1;  // lanes 0-15 -> K=k0,k0+1 ; lanes 16-31 -> k0+2,k0+3

  const v8f vzero = {0.f, 0.f, 0.f, 0.f, 0.f, 0.f, 0.f, 0.f};
  v8f acc[2][2];
  acc[0][0] = vzero; acc[0][1] = vzero; acc[1][0] = vzero; acc[1][1] = vzero;

#if !USE_TDM
  float4 rx[4], rw[4];  // register staging (fallback path)
#endif

#if USE_TDM
#define STAGE_ISSUE(bufi, kk0)                                                   \
  do {                                                                           \
    if (wave == 0) {                                                             \
      tdm_load_2d(ldsOff(&sX[bufi][0]), X + (size_t)row0 * DIM + (kk0), 32, 64,  \
                  DIM, 4, 3); /* 128B rows, +4 DW pad -> stride 36 */            \
      tdm_load_2d(ldsOff(&sW[bufi][0]), W + (size_t)(kk0)*DIM + col0, 64, 32,    \
                  DIM, 5, 15); /* 256B rows, +16 DW pad -> stride 80 */          \
    }                                                                            \
  } while (0)
#define STAGE_COMMIT(bufi)                                                       \
  do {                                                                           \
    if (wave == 0) __builtin_amdgcn_s_wait_tensorcnt(0);                         \
  } while (0)
#else
#define STAGE_ISSUE(bufi, kk0)                                                   \
  do {                                                                           \
    _Pragma("unroll") for (int i = 0; i < 4; ++i) {                              \
      const int f = t + (i << 7);                                                \
      rx[i] = *reinterpret_cast<const float4*>(                                  \
          X + (size_t)(row0 + (f >> 3)) * DIM + (kk0) + ((f & 7) << 2));         \
      rw[i] = *reinterpret_cast<const float4*>(                                  \
          W + (size_t)((kk0) + (f >> 4)) * DIM + col0 + ((f & 15) << 2));        \
    }                                                                            \
  } while (0)
#define STAGE_COMMIT(bufi)                                                       \
  do {                                                                           \
    _Pragma("unroll") for (int i = 0; i < 4; ++i) {                              \
      const int f = t + (i << 7);                                                \
      *reinterpret_cast<float4*>(                                                \
          &sX[bufi][(f >> 3) * XSTRIDE + ((f & 7) << 2)]) = rx[i];               \
      *reinterpret_cast<float4*>(                                                \
          &sW[bufi][(f >> 4) * WSTRIDE + ((f & 15) << 2)]) = rw[i];              \
    }                                                                            \
  } while (0)
#endif

  // prologue: stage slab 0
  STAGE_ISSUE(0, 0);
  STAGE_COMMIT(0);
  __syncthreads();

  for (int k0 = 0; k0 < DIM; k0 += 32) {
    const int buf = (k0 >> 5) & 1;
    const bool more = (k0 + 32) < DIM;
    if (more) STAGE_ISSUE(buf ^ 1, k0 + 32);  // overlap DMA with WMMAs below

    const float* sXb = &sX[buf][0];
    const float* sWb = &sW[buf][0];
#pragma unroll
    for (int kk = 0; kk < 32; kk += 4) {
      const v2f a0 = *reinterpret_cast<const v2f*>(sXb + (wm + lr) * XSTRIDE + kk + ksel);
      const v2f a1 = *reinterpret_cast<const v2f*>(sXb + (wm + 16 + lr) * XSTRIDE + kk + ksel);
      const int kr = (kk + ksel) * WSTRIDE;
      const v2f b0 = {sWb[kr + wn + lr], sWb[kr + WSTRIDE + wn + lr]};
      const v2f b1 = {sWb[kr + wn + 16 + lr], sWb[kr + WSTRIDE + wn + 16 + lr]};
      acc[0][0] = wmma_f32_k4(a0, b0, acc[0][0]);
      acc[0][1] = wmma_f32_k4(a0, b1, acc[0][1]);
      acc[1][0] = wmma_f32_k4(a1, b0, acc[1][0]);
      acc[1][1] = wmma_f32_k4(a1, b1, acc[1][1]);
    }

    if (more) STAGE_COMMIT(buf ^ 1);
    __syncthreads();
  }

  // C/D layout: VGPR r holds M = r (lanes 0-15) or r+8 (lanes 16-31), N = lr.
#pragma unroll
  for (int i = 0; i < 2; ++i) {
#pragma unroll
    for (int j = 0; j < 2; ++j) {
      const int n = col0 + wn + j * 16 + lr;
#pragma unroll
      for (int r = 0; r < 8; ++r) {
        const int m = row0 + wm + i * 16 + r + hi * 8;
        if (transposeOut)
          Out[(size_t)n * SEQ + m] = acc[i][j][r];
        else
          Out[(size_t)m * DIM + n] = acc[i][j][r];
      }
    }
  }
}

// ---------------------------------------------------------------------------
// Flash attention: block = 128 threads (4 waves) handles 16 query rows.
// Per 16-key chunk: scores split 256-features-per-wave (64 WMMAs), merged via
// ds_add_f32 atomics into a 16x16 LDS tile; online softmax by 16 threads;
// each wave rescales its private 16x256 output slice and accumulates
// P(16x16) x V^T chunk with 64 WMMAs. No 64MB score matrix ever materialized.
// ---------------------------------------------------------------------------
__global__ __launch_bounds__(128) void attn_fwd(const float* __restrict__ Q,
                                                const float* __restrict__ K,
                                                const float* __restrict__ Vt,
                                                float* __restrict__ Out) {
  __shared__ __align__(16) float sS[256];  // 16x16 scores / P
  __shared__ float sM[16];                 // running row max
  __shared__ float sL[16];                 // running row sum
  __shared__ float sA[16];                 // per-row rescale factor

  const int t = threadIdx.x;
  const int lane = t & 31;
  const int wave = t >> 5;
  const int lr = lane & 15;
  const int hi = lane >> 4;
  const int ksel = hi << 1;
  const int q0 = blockIdx.x << 4;
  const int col0 = wave << 8;   // 256 output cols per wave
  const int kbase = wave << 8;  // 256 reduction features per wave (scores)

  const v8f vzero = {0.f, 0.f, 0.f, 0.f, 0.f, 0.f, 0.f, 0.f};
  v8f o[16];
#pragma unroll
  for (int i = 0; i < 16; ++i) o[i] = vzero;

  if (t < 16) {
    sM[t] = -__builtin_inff();
    sL[t] = 0.f;
  }

  const float* qrow = Q + (size_t)(q0 + lr) * DIM + kbase;

  for (int key0 = 0; key0 < SEQ; key0 += 16) {
    sS[t] = 0.f;
    sS[t + 128] = 0.f;
    __syncthreads();

    // --- partial scores over this wave's 256-feature slice ---
    const float* krow = K + (size_t)(key0 + lr) * DIM + kbase;
    v8f sc = vzero;
#pragma unroll
    for (int kk = 0; kk < 256; kk += 4) {
      const v2f a = *reinterpret_cast<const v2f*>(qrow + kk + ksel);
      const v2f b = *reinterpret_cast<const v2f*>(krow + kk + ksel);
      sc = wmma_f32_k4(a, b, sc);
    }
#pragma unroll
    for (int r = 0; r < 8; ++r)
      atomicAdd(&sS[(r + hi * 8) * 16 + lr], sc[r] * 0.03125f);  // 1/sqrt(1024)
    __syncthreads();

    // --- online softmax update (one thread per query row) ---
    if (t < 16) {
      const float mOld = sM[t];
      float mNew = mOld;
#pragma unroll
      for (int c = 0; c < 16; ++c) mNew = fmaxf(mNew, sS[t * 16 + c]);
      const float alpha = __expf(mOld - mNew);  // exp(-inf)=0 on first chunk
      float sum = 0.f;
#pragma unroll
      for (int c = 0; c < 16; ++c) {
        const float p = __expf(sS[t * 16 + c] - mNew);
        sS[t * 16 + c] = p;
        sum += p;
      }
      sL[t] = sL[t] * alpha + sum;
      sM[t] = mNew;
      sA[t] = alpha;
    }
    __syncthreads();

    // --- rescale accumulators, then O += P x V^T-chunk ---
    float al[8];
#pragma unroll
    for (int r = 0; r < 8; ++r) al[r] = sA[r + hi * 8];
#pragma unroll
    for (int i = 0; i < 16; ++i)
#pragma unroll
      for (int r = 0; r < 8; ++r) o[i][r] *= al[r];

    v2f pa[4];  // A fragments of P (same for all 16 N tiles)
#pragma unroll
    for (int kk = 0; kk < 4; ++kk)
      pa[kk] = *reinterpret_cast<const v2f*>(&sS[lr * 16 + kk * 4 + ksel]);

#pragma unroll
    for (int i = 0; i < 16; ++i) {
      const float* vcol = Vt + (size_t)(col0 + i * 16 + lr) * SEQ + key0;
#pragma unroll
      for (int kk = 0; kk < 4; ++kk) {
        const v2f b = *reinterpret_cast<const v2f*>(vcol + kk * 4 + ksel);
        o[i] = wmma_f32_k4(pa[kk], b, o[i]);
      }
    }
    __syncthreads();  // protect sS before next chunk re-zeroes it
  }

  // --- normalize by row sums and store ---
  float inv[8];
#pragma unroll
  for (int r = 0; r < 8; ++r) inv[r] = 1.0f / sL[r + hi * 8];
#pragma unroll
  for (int i = 0; i < 16; ++i) {
    const int n = col0 + i * 16 + lr;
#pragma unroll
    for (int r = 0; r < 8; ++r)
      Out[(size_t)(q0 + r + hi * 8) * DIM + n] = o[i][r] * inv[r];
  }
}

extern "C" void kernel_launch(void* const* d_in, const int* in_sizes, int n_in,
                              void* d_out, int out_size, void* d_ws,
                              size_t ws_size, hipStream_t stream) {
  (void)in_sizes; (void)n_in; (void)out_size; (void)ws_size;
  const float* x  = (const float*)d_in[0];
  const float* Wq = (const float*)d_in[1];
  const float* Wk = (const float*)d_in[2];
  const float* Wv = (const float*)d_in[3];

  float* Qm = (float*)d_ws;                  // [SEQ][DIM]
  float* Km = Qm + (size_t)SEQ * DIM;        // [SEQ][DIM]
  float* Vt = Km + (size_t)SEQ * DIM;        // [DIM][SEQ]  (transposed V)

  dim3 gg(DIM / 64, SEQ / 64);
  qkv_gemm<<<gg, 128, 0, stream>>>(x, Wq, Qm, 0);
  qkv_gemm<<<gg, 128, 0, stream>>>(x, Wk, Km, 0);
  qkv_gemm<<<gg, 128, 0, stream>>>(x, Wv, Vt, 1);

  attn_fwd<<<SEQ / 16, 128, 0, stream>>>(Qm, Km, Vt, (float*)d_out);
}